// Group_44547400794275
// MI455X (gfx1250) — compile-verified
//
#include <hip/hip_runtime.h>
#include <hip/hip_bf16.h>

typedef __attribute__((ext_vector_type(2))) float v2f;
typedef __attribute__((ext_vector_type(8))) float v8f;

#define NB      32      // batches
#define NPTS    8192    // points per batch
#define NG      256     // num_group
#define GS      32      // group_size
#define GT      16      // groups per workgroup (one WMMA tile of rows)
#define CHUNK   512     // points processed per LDS tile pass
#define TSTRIDE (CHUNK + 4)

// ---------------------------------------------------------------------------
// Kernel 1: farthest point sampling. One block per batch, 1024 threads.
// Writes centers (xyz of selected indices) into the center region of d_out.
// ---------------------------------------------------------------------------
__global__ __launch_bounds__(1024) void fps_kernel(const float* __restrict__ xyz,
                                                   float* __restrict__ centers)
{
    const int b = blockIdx.x;
    const int t = threadIdx.x;
    const int lane = t & 31;
    const int wave = t >> 5;
    const float* p = xyz + (size_t)b * NPTS * 3;

    __shared__ float sCx, sCy, sCz;
    __shared__ int   sFar;
    __shared__ float sVal[32];
    __shared__ int   sIdx[32];

    float dist[8];
#pragma unroll
    for (int r = 0; r < 8; ++r) dist[r] = 1e10f;

    if (t == 0) sFar = 0;
    __syncthreads();

    for (int i = 0; i < NG; ++i) {
        if (t == 0) {
            int f = sFar;
            float cx = p[f * 3 + 0], cy = p[f * 3 + 1], cz = p[f * 3 + 2];
            sCx = cx; sCy = cy; sCz = cz;
            int o = (b * NG + i) * 3;
            centers[o + 0] = cx; centers[o + 1] = cy; centers[o + 2] = cz;
        }
        __syncthreads();
        const float cx = sCx, cy = sCy, cz = sCz;

        float bestV = -1.0f;
        int   bestI = 0x7fffffff;
#pragma unroll
        for (int r = 0; r < 8; ++r) {
            int pt = t + 1024 * r;
            float dx = p[pt * 3 + 0] - cx;
            float dy = p[pt * 3 + 1] - cy;
            float dz = p[pt * 3 + 2] - cz;
            float d  = dx * dx + dy * dy + dz * dz;
            float dd = fminf(dist[r], d);
            dist[r] = dd;
            if (dd > bestV) { bestV = dd; bestI = pt; }   // pt increases with r
        }
        // wave32 argmax, lowest-index tie-break (matches jnp.argmax)
        for (int off = 16; off > 0; off >>= 1) {
            float ov = __shfl_xor(bestV, off, 32);
            int   oi = __shfl_xor(bestI, off, 32);
            if (ov > bestV || (ov == bestV && oi < bestI)) { bestV = ov; bestI = oi; }
        }
        if (lane == 0) { sVal[wave] = bestV; sIdx[wave] = bestI; }
        __syncthreads();
        if (wave == 0) {
            float v = sVal[lane];
            int  ix = sIdx[lane];
            for (int off = 16; off > 0; off >>= 1) {
                float ov = __shfl_xor(v, off, 32);
                int   oi = __shfl_xor(ix, off, 32);
                if (ov > v || (ov == v && oi < ix)) { v = ov; ix = oi; }
            }
            if (lane == 0) sFar = ix;
        }
        __syncthreads();
    }
}

// ---------------------------------------------------------------------------
// Kernel 2: distances via WMMA f32 16x16x4 + streaming top-32 + bitonic sort.
// One 512-thread workgroup per (batch, 16-group tile). Wave w owns group w.
// ---------------------------------------------------------------------------
__global__ __launch_bounds__(512) void group_kernel(const float* __restrict__ xyz,
                                                    const float* __restrict__ centers,
                                                    float* __restrict__ outN)
{
    const int blk   = blockIdx.x;
    const int b     = blk >> 4;          // 16 tiles per batch
    const int g0    = (blk & 15) * GT;
    const int t     = threadIdx.x;
    const int lane  = t & 31;
    const int w     = t >> 5;            // wave id == group row in this tile

    const float* pts = xyz + (size_t)b * NPTS * 3;

    __shared__ float tile[GT][TSTRIDE];

    // Build A fragment (16x4 f32): row m = [-2cx, -2cy, -2cz, 1]
    // ISA layout: lanes 0-15 hold K={0,1}, lanes 16-31 hold K={2,3}, M = lane%16
    {
    }
    const int mA = lane & 15;
    const float* cA = centers + (size_t)(b * NG + g0 + mA) * 3;
    float cax = cA[0], cay = cA[1], caz = cA[2];
    v2f afrag;
    if (lane < 16) { afrag.x = -2.0f * cax; afrag.y = -2.0f * cay; }
    else           { afrag.x = -2.0f * caz; afrag.y = 1.0f; }

    // register-resident top-32: one (dist, idx) entry per lane
    float hd = 1e30f;
    int   hi = 0x40000000 + lane;        // distinct keys -> unique argmax
    float curMax  = 1e30f;
    int   maxLane = 31;

    for (int base = 0; base < NPTS; base += CHUNK) {
        // ---- compute phase: 32 16x16 tiles per chunk, 2 per wave ----
#pragma unroll
        for (int tt = 0; tt < 2; ++tt) {
            const int tl = w * 2 + tt;
            const int colLocal = tl * 16 + (lane & 15);
            const int n = base + colLocal;
            const float* q = pts + (size_t)n * 3;
            float x = q[0], y = q[1], z = q[2];
            float sq = x * x + y * y + z * z;
            // B fragment (4x16 f32): rows [x; y; z; ||x||^2]
            v2f bfrag;
            if (lane < 16) { bfrag.x = x; bfrag.y = y; }
            else           { bfrag.x = z; bfrag.y = sq; }
            v8f cacc = {};
            v8f dacc = __builtin_amdgcn_wmma_f32_16x16x4_f32(
                false, afrag, false, bfrag, (short)0, cacc, false, false);
            const int half = (lane >> 4) << 3;  // 0 or 8
#pragma unroll
            for (int r = 0; r < 8; ++r)
                tile[r + half][colLocal] = dacc[r];
        }
        __syncthreads();

        // ---- merge phase: wave w streams its row into its top-32 ----
        for (int j = 0; j < CHUNK; j += 32) {
            float dv = tile[w][j + lane];
            int   gi = base + j + lane;
            unsigned long long msk = __ballot(dv < curMax);
            while (msk) {
                int src = __ffsll(msk) - 1;
                msk &= msk - 1;
                float dc = __shfl(dv, src, 32);
                int   ic = __shfl(gi, src, 32);
                if (dc < curMax) {
                    if (lane == maxLane) { hd = dc; hi = ic; }
                    // recompute (max dist, max idx) entry -> evict slot
                    float mv = hd; int mi = hi; int ml = lane;
                    for (int off = 16; off > 0; off >>= 1) {
                        float ov = __shfl_xor(mv, off, 32);
                        int   oi = __shfl_xor(mi, off, 32);
                        int   ol = __shfl_xor(ml, off, 32);
                        if (ov > mv || (ov == mv && oi > mi)) { mv = ov; mi = oi; ml = ol; }
                    }
                    curMax = mv; maxLane = ml;
                }
            }
        }
        __syncthreads();
    }

    // ---- bitonic sort the 32 entries ascending by (dist, idx) ----
    float d = hd;
    int  ix = hi;
#pragma unroll
    for (int k = 2; k <= 32; k <<= 1) {
#pragma unroll
        for (int j = k >> 1; j > 0; j >>= 1) {
            float od = __shfl_xor(d, j, 32);
            int   oi = __shfl_xor(ix, j, 32);
            bool up      = ((lane & k) == 0);
            bool lower   = ((lane & j) == 0);
            bool takeMin = (lower == up);
            bool oLess   = (od < d) || (od == d && oi < ix);
            if (oLess == takeMin) { d = od; ix = oi; }
        }
    }

    // ---- write neighborhood: xyz[idx] - center ----
    const int g = g0 + w;
    const float* cw = centers + (size_t)(b * NG + g) * 3;
    const float* q  = pts + (size_t)ix * 3;
    float* o = outN + ((size_t)(b * NG + g) * GS + lane) * 3;
    o[0] = q[0] - cw[0];
    o[1] = q[1] - cw[1];
    o[2] = q[2] - cw[2];
}

extern "C" void kernel_launch(void* const* d_in, const int* in_sizes, int n_in,
                              void* d_out, int out_size, void* d_ws, size_t ws_size,
                              hipStream_t stream) {
    (void)in_sizes; (void)n_in; (void)d_ws; (void)ws_size; (void)out_size;
    const float* xyz = (const float*)d_in[0];
    float* outN    = (float*)d_out;                              // [B,256,32,3]
    float* centers = (float*)d_out + (size_t)NB * NG * GS * 3;   // [B,256,3]

    fps_kernel<<<NB, 1024, 0, stream>>>(xyz, centers);
    group_kernel<<<NB * (NG / GT), 512, 0, stream>>>(xyz, centers, outN);
}